// BarrierNetLayer_70824010711242
// MI455X (gfx1250) — compile-verified
//
#include <hip/hip_runtime.h>

// Batched HOCBF safety-filter QP via fixed-iteration primal-dual IPM.
// One lane (wave32) solves one QP entirely in registers.
//   min 0.5||u - u_ref||^2  s.t.  A_cbf u <= b_cbf,  -10 <= u <= 10
// G = [A_cbf; I; -I] (16x4), h = [b_cbf; 10*1; 10*1].
//
// WMMA note: G is per-batch, so all matrix algebra here is block-diagonal
// over the batch dimension and does not factor into any 16x16xK GEMM;
// per-lane scalar f32 VALU is the optimal mapping on gfx1250 for this shape.

namespace {

constexpr int   kM     = 4;     // control dims
constexpr int   kNC    = 8;     // CBF rows
constexpr int   kN     = 16;    // total inequality rows
constexpr int   kIters = 20;
constexpr float kSigma = 0.1f;
constexpr float kBound = 10.0f; // U_HI = -U_LO

// raw v_rcp_f32: ~1 ulp, same order as FMA rounding; fine for elementwise use.
__device__ __forceinline__ float rcp_raw(float x) {
  return __builtin_amdgcn_rcpf(x);
}
// rcp + one Newton-Raphson step (~0.5 ulp) for the LDL^T pivots.
__device__ __forceinline__ float rcp_nr(float x) {
  float r = __builtin_amdgcn_rcpf(x);
  r = r * __builtin_fmaf(-x, r, 2.0f);
  return r;
}

__global__ __launch_bounds__(256) void hocbf_qp_ipm_kernel(
    const float* __restrict__ u_ref,   // [B,4,1]
    const float* __restrict__ A_cbf,   // [B,8,4]
    const float* __restrict__ b_cbf,   // [B,8,1]
    float* __restrict__ out,           // [B,4,1]
    int nbatch)
{
  const int b = blockIdx.x * blockDim.x + threadIdx.x;
  if (b >= nbatch) return;

  // gfx1250 prefetch of this lane's constraint matrix (global_prefetch_b8),
  // temporal locality -> WGP-scope prefetch.
  __builtin_prefetch(A_cbf + (size_t)b * (kNC * kM), 0, 3);

  // ---------------- load problem data (b128 per lane) ----------------
  const float4 urv = *(const float4*)(u_ref + (size_t)b * kM);
  const float q[kM] = {-urv.x, -urv.y, -urv.z, -urv.w};

  float a[kNC][kM];
#pragma unroll
  for (int n = 0; n < kNC; ++n) {
    float4 r = *(const float4*)(A_cbf + (size_t)b * (kNC * kM) + n * kM);
    a[n][0] = r.x; a[n][1] = r.y; a[n][2] = r.z; a[n][3] = r.w;
  }

  float h[kN];
  {
    float4 b0 = *(const float4*)(b_cbf + (size_t)b * kNC);
    float4 b1 = *(const float4*)(b_cbf + (size_t)b * kNC + 4);
    h[0] = b0.x; h[1] = b0.y; h[2] = b0.z; h[3] = b0.w;
    h[4] = b1.x; h[5] = b1.y; h[6] = b1.z; h[7] = b1.w;
#pragma unroll
    for (int i = 8; i < kN; ++i) h[i] = kBound;
  }

  // ---------------- IPM state ----------------
  float u[kM] = {0.f, 0.f, 0.f, 0.f};
  float s[kN], z[kN];
#pragma unroll
  for (int n = 0; n < kN; ++n) { s[n] = h[n]; z[n] = 1.0f; }

#pragma unroll 1
  for (int it = 0; it < kIters; ++it) {
    // ---- Gu (rows 8..15 are +/-I, so only 8 dense rows) ----
    float Gu[kN];
#pragma unroll
    for (int n = 0; n < kNC; ++n)
      Gu[n] = a[n][0]*u[0] + a[n][1]*u[1] + a[n][2]*u[2] + a[n][3]*u[3];
#pragma unroll
    for (int p = 0; p < kM; ++p) { Gu[8 + p] = u[p]; Gu[12 + p] = -u[p]; }

    // ---- residuals, duality measure ----
    float rp[kN];
    float szsum = 0.f;
#pragma unroll
    for (int n = 0; n < kN; ++n) {
      rp[n] = Gu[n] + s[n] - h[n];
      szsum += s[n] * z[n];
    }
    const float smu = kSigma * (szsum * (1.0f / 16.0f)); // sigma * mu

    // inv_s, d = z/s,  t = r_c/s + d*r_p = sigma*mu/s - z + d*r_p
    float inv_s[kN], d[kN], t[kN];
#pragma unroll
    for (int n = 0; n < kN; ++n) {
      inv_s[n] = rcp_raw(s[n]);
      d[n] = z[n] * inv_s[n];
      t[n] = __builtin_fmaf(smu, inv_s[n], -z[n]) + d[n] * rp[n];
    }

    // ---- rhs = -(r_d + G^T t),  r_d = u + q + G^T z ----
    float rhs[kM];
#pragma unroll
    for (int p = 0; p < kM; ++p) {
      float gtz = z[8 + p] - z[12 + p];
      float gtt = t[8 + p] - t[12 + p];
#pragma unroll
      for (int n = 0; n < kNC; ++n) {
        gtz = __builtin_fmaf(a[n][p], z[n], gtz);
        gtt = __builtin_fmaf(a[n][p], t[n], gtt);
      }
      rhs[p] = -(u[p] + q[p] + gtz + gtt);
    }

    // ---- K = I + G^T diag(d) G  (SPD 4x4; box rows -> diagonal) ----
    float K00 = 1.f + d[8]  + d[12];
    float K11 = 1.f + d[9]  + d[13];
    float K22 = 1.f + d[10] + d[14];
    float K33 = 1.f + d[11] + d[15];
    float K10 = 0.f, K20 = 0.f, K21 = 0.f, K30 = 0.f, K31 = 0.f, K32 = 0.f;
#pragma unroll
    for (int n = 0; n < kNC; ++n) {
      const float w0 = d[n]*a[n][0], w1 = d[n]*a[n][1];
      const float w2 = d[n]*a[n][2], w3 = d[n]*a[n][3];
      K00 = __builtin_fmaf(w0, a[n][0], K00);
      K10 = __builtin_fmaf(w1, a[n][0], K10);
      K11 = __builtin_fmaf(w1, a[n][1], K11);
      K20 = __builtin_fmaf(w2, a[n][0], K20);
      K21 = __builtin_fmaf(w2, a[n][1], K21);
      K22 = __builtin_fmaf(w2, a[n][2], K22);
      K30 = __builtin_fmaf(w3, a[n][0], K30);
      K31 = __builtin_fmaf(w3, a[n][1], K31);
      K32 = __builtin_fmaf(w3, a[n][2], K32);
      K33 = __builtin_fmaf(w3, a[n][3], K33);
    }

    // ---- branch-free LDL^T factorization + solve: K du = rhs ----
    float du[kM];
    {
      const float iD0 = rcp_nr(K00);
      const float L10 = K10 * iD0, L20 = K20 * iD0, L30 = K30 * iD0;
      const float D1  = K11 - L10 * K10;
      const float iD1 = rcp_nr(D1);
      const float L21 = (K21 - L20 * K10) * iD1;
      const float L31 = (K31 - L30 * K10) * iD1;
      const float D2  = K22 - L20 * K20 - L21 * L21 * D1;
      const float iD2 = rcp_nr(D2);
      const float L32 = (K32 - L30 * K20 - L31 * L21 * D1) * iD2;
      const float D3  = K33 - L30 * K30 - L31 * L31 * D1 - L32 * L32 * D2;
      const float iD3 = rcp_nr(D3);
      const float y0 = rhs[0];
      const float y1 = rhs[1] - L10 * y0;
      const float y2 = rhs[2] - L20 * y0 - L21 * y1;
      const float y3 = rhs[3] - L30 * y0 - L31 * y1 - L32 * y2;
      const float x3 = y3 * iD3;
      const float x2 = y2 * iD2 - L32 * x3;
      const float x1 = y1 * iD1 - L21 * x2 - L31 * x3;
      const float x0 = y0 * iD0 - L10 * x1 - L20 * x2 - L30 * x3;
      du[0] = x0; du[1] = x1; du[2] = x2; du[3] = x3;
    }

    // ---- G du ----
    float Gdu[kN];
#pragma unroll
    for (int n = 0; n < kNC; ++n)
      Gdu[n] = a[n][0]*du[0] + a[n][1]*du[1] + a[n][2]*du[2] + a[n][3]*du[3];
#pragma unroll
    for (int p = 0; p < kM; ++p) { Gdu[8 + p] = du[p]; Gdu[12 + p] = -du[p]; }

    // ---- directions + fraction-to-boundary via inverted reduction:
    //      min_n(v/-dv) == 1 / max_n(-dv/v); v in {s,z} > 0.
    //      Uses inv_s already computed; only 17 extra rcps instead of 32. ----
    float dz[kN], dsv[kN];
    float mratio = 0.f;
#pragma unroll
    for (int n = 0; n < kN; ++n) {
      dz[n]  = __builtin_fmaf(d[n], Gdu[n], t[n]);   // t + d*Gdu
      dsv[n] = -rp[n] - Gdu[n];
      const float ws = (dsv[n] < 0.f) ? (-dsv[n]) * inv_s[n] : 0.f;
      const float wz = (dz[n]  < 0.f) ? (-dz[n])  * rcp_raw(z[n]) : 0.f;
      mratio = fmaxf(mratio, fmaxf(ws, wz));
    }
    // mratio==0 -> rcp gives +inf -> alpha = 1 (matches the 1e9 "big" path).
    const float alpha = fminf(1.0f, 0.99f * rcp_raw(mratio));

#pragma unroll
    for (int p = 0; p < kM; ++p) u[p] = __builtin_fmaf(alpha, du[p], u[p]);
#pragma unroll
    for (int n = 0; n < kN; ++n) {
      s[n] = __builtin_fmaf(alpha, dsv[n], s[n]);
      z[n] = __builtin_fmaf(alpha, dz[n], z[n]);
    }
  }

  float4 o;
  o.x = u[0]; o.y = u[1]; o.z = u[2]; o.w = u[3];
  *(float4*)(out + (size_t)b * kM) = o;
}

} // namespace

extern "C" void kernel_launch(void* const* d_in, const int* in_sizes, int n_in,
                              void* d_out, int out_size, void* d_ws, size_t ws_size,
                              hipStream_t stream) {
  const float* u_ref = (const float*)d_in[0]; // [B,4,1]
  const float* A_cbf = (const float*)d_in[1]; // [B,8,4]
  const float* b_cbf = (const float*)d_in[2]; // [B,8,1]
  float* out = (float*)d_out;                 // [B,4,1]

  const int nbatch = in_sizes[0] / 4;
  const int block = 256;                      // 8 wave32s per workgroup
  const int grid  = (nbatch + block - 1) / block;
  hocbf_qp_ipm_kernel<<<grid, block, 0, stream>>>(u_ref, A_cbf, b_cbf, out, nbatch);
}